// CAPUDFNetwork_52802327937041
// MI455X (gfx1250) — compile-verified
//
#include <hip/hip_runtime.h>

// Closed-form masked box-boundary distance (SIZE = 0.4), fused:
//   q = |x| - SIZE
//   inside (all q<0): d = -max(q)
//   outside:          d = sqrt(sum(max(q,0)^2))
//
// Memory-bound: 16 B/point single-pass => use 128-bit NT loads/stores.

typedef float v4f __attribute__((ext_vector_type(4)));

#define BOX_HALF 0.4f

__device__ __forceinline__ float box_udf(float x, float y, float z) {
    float qx = __builtin_fabsf(x) - BOX_HALF;
    float qy = __builtin_fabsf(y) - BOX_HALF;
    float qz = __builtin_fabsf(z) - BOX_HALF;
    float rx = __builtin_fmaxf(qx, 0.0f);
    float ry = __builtin_fmaxf(qy, 0.0f);
    float rz = __builtin_fmaxf(qz, 0.0f);
    float sumsq = __builtin_fmaf(rx, rx, __builtin_fmaf(ry, ry, rz * rz));
    float d_out = __builtin_sqrtf(sumsq);
    float qmax  = __builtin_fmaxf(qx, __builtin_fmaxf(qy, qz));
    // strict inside test (all q < 0) matches reference's jnp.all(q < 0)
    return (qmax < 0.0f) ? -qmax : d_out;
}

// Each thread: 4 points = 12 floats = 3 x b128 loads (16B-aligned: base = 48*t),
// 1 x b128 store. Non-temporal hints on the streaming traffic.
__global__ __launch_bounds__(256) void box_udf_kernel(
    const float* __restrict__ in, float* __restrict__ out, int npts) {
    const int i    = blockIdx.x * blockDim.x + threadIdx.x;  // vec4-of-points index
    const int nvec = npts >> 2;

    if (i < nvec) {
        const v4f* __restrict__ in4 = (const v4f*)in;
        v4f a = __builtin_nontemporal_load(&in4[3 * i + 0]);  // p0.xyz, p1.x
        v4f b = __builtin_nontemporal_load(&in4[3 * i + 1]);  // p1.yz, p2.xy
        v4f c = __builtin_nontemporal_load(&in4[3 * i + 2]);  // p2.z, p3.xyz

        v4f r;
        r.x = box_udf(a.x, a.y, a.z);
        r.y = box_udf(a.w, b.x, b.y);
        r.z = box_udf(b.z, b.w, c.x);
        r.w = box_udf(c.y, c.z, c.w);

        __builtin_nontemporal_store(r, &((v4f*)out)[i]);
    }

    // Tail (npts % 4 != 0): at most 3 points, done by thread 0 of block 0.
    // (N = 8388608 is divisible by 4, so this is dead code in practice.)
    if (blockIdx.x == 0 && threadIdx.x == 0) {
        for (int p = nvec << 2; p < npts; ++p) {
            out[p] = box_udf(in[3 * p + 0], in[3 * p + 1], in[3 * p + 2]);
        }
    }
}

extern "C" void kernel_launch(void* const* d_in, const int* in_sizes, int n_in,
                              void* d_out, int out_size, void* d_ws, size_t ws_size,
                              hipStream_t stream) {
    const float* in = (const float*)d_in[0];
    float* out      = (float*)d_out;

    const int npts    = in_sizes[0] / 3;   // inputs is [N,3] float32
    const int nvec    = npts >> 2;
    const int threads = 256;               // 8 wave32 waves per block
    int blocks        = (nvec + threads - 1) / threads;
    if (blocks < 1) blocks = 1;            // still run tail path for tiny N

    box_udf_kernel<<<blocks, threads, 0, stream>>>(in, out, npts);
}